// LEAP_74371653697613
// MI455X (gfx1250) — compile-verified
//
#include <hip/hip_runtime.h>
#include <math.h>

#define E      1024
#define NCTX   320
#define TSTEPS 65
#define VOUT   32002
#define V0TOK  32000
#define MPAD   80

typedef __attribute__((ext_vector_type(2))) float v2f;
typedef __attribute__((ext_vector_type(8))) float v8f;

__device__ __forceinline__ float waveSum(float v) {
#pragma unroll
    for (int o = 16; o > 0; o >>= 1) v += __shfl_down(v, o, 32);
    return v;
}
__device__ __forceinline__ float waveMax(float v) {
#pragma unroll
    for (int o = 16; o > 0; o >>= 1) v = fmaxf(v, __shfl_down(v, o, 32));
    return v;
}
__device__ __forceinline__ float blockSum(float v, float* red) {
    const int lane = threadIdx.x & 31, wv = threadIdx.x >> 5;
    const int nw = (int)((blockDim.x + 31) >> 5);
    v = waveSum(v);
    __syncthreads();
    if (lane == 0) red[wv] = v;
    __syncthreads();
    if (wv == 0) {
        float q = (lane < nw) ? red[lane] : 0.f;
        q = waveSum(q);
        if (lane == 0) red[0] = q;
    }
    __syncthreads();
    float r = red[0];
    __syncthreads();
    return r;
}
__device__ __forceinline__ float blockMax(float v, float* red) {
    const int lane = threadIdx.x & 31, wv = threadIdx.x >> 5;
    const int nw = (int)((blockDim.x + 31) >> 5);
    v = waveMax(v);
    __syncthreads();
    if (lane == 0) red[wv] = v;
    __syncthreads();
    if (wv == 0) {
        float q = (lane < nw) ? red[lane] : -__builtin_inff();
        q = waveMax(q);
        if (lane == 0) red[0] = q;
    }
    __syncthreads();
    float r = red[0];
    __syncthreads();
    return r;
}

// Gather ctx / dec_x embeddings, precompute ctx_score = ctx @ w_e + b, init h0,
// zero the M-padding rows of the relu(h) matrix used by the final WMMA GEMM.
__global__ void LEAP_setup(const int* __restrict__ diag, const int* __restrict__ proc,
                           const int* __restrict__ med, const int* __restrict__ dtok,
                           const float* __restrict__ enc_emb, const float* __restrict__ dec_emb,
                           const float* __restrict__ attn_w, const float* __restrict__ attn_b,
                           float* __restrict__ ctx, float* __restrict__ ctx_score,
                           float* __restrict__ dec_x, float* __restrict__ h,
                           float* __restrict__ Hrelu) {
    __shared__ float red[8];
    const int b = blockIdx.x, tid = threadIdx.x;
    if (b < NCTX) {
        const int srcRow = (b < 128) ? diag[b] : (b < 192 ? proc[b - 128] : med[b - 192]);
        const float* src = enc_emb + (size_t)srcRow * E;
        const float* w_e = attn_w + E;
        float* dst = ctx + (size_t)b * E;
        float p = 0.f;
        for (int i = tid; i < E; i += 256) { float v = src[i]; dst[i] = v; p += v * w_e[i]; }
        p = blockSum(p, red);
        if (tid == 0) ctx_score[b] = p + attn_b[0];
    } else if (b < NCTX + TSTEPS) {
        const int t = b - NCTX;
        const int tok = (t == 0) ? V0TOK : dtok[t - 1];
        const float* src = dec_emb + (size_t)tok * E;
        float* dst = dec_x + (size_t)t * E;
        for (int i = tid; i < E; i += 256) { float v = src[i]; dst[i] = v; if (t == 0) h[i] = v; }
    } else {
        const int row = TSTEPS + (b - (NCTX + TSTEPS));
        float* dst = Hrelu + (size_t)row * E;
        for (int i = tid; i < E; i += 256) dst[i] = 0.f;
    }
}

// Gate combine (h_new, relu capture) fused with next-step attention softmax.
// init==1: skip gates, just compute softmax from current h (step 0 bootstrap).
__global__ void LEAP_combine_attn(const float* __restrict__ gi, const float* __restrict__ gh,
                                  float* __restrict__ h, const float* __restrict__ attn_w,
                                  const float* __restrict__ ctx_score, float* __restrict__ a,
                                  float* __restrict__ Hrelu, int t, int init) {
    __shared__ float red[32];
    const int j = threadIdx.x;   // 1024 threads
    float hn;
    if (!init) {
        const float r = 1.f / (1.f + expf(-(gi[j] + gh[j])));
        const float z = 1.f / (1.f + expf(-(gi[E + j] + gh[E + j])));
        const float n = tanhf(gi[2 * E + j] + r * gh[2 * E + j]);
        const float hp = h[j];
        hn = (1.f - z) * n + z * hp;
        h[j] = hn;
        Hrelu[(size_t)t * E + j] = fmaxf(hn, 0.f);
    } else {
        hn = h[j];
    }
    // d = h . w_h  (w_h = attn_w[0:E])
    const float d = blockSum(hn * attn_w[j], red);
    const float sc = (j < NCTX) ? ctx_score[j] + d : -__builtin_inff();
    const float mx = blockMax(sc, red);
    const float e = (j < NCTX) ? expf(sc - mx) : 0.f;
    const float s = blockSum(e, red);
    if (j < NCTX) a[j] = e / s;
}

// c = a @ ctx  (320 x 1024 weighted sum); attention weights staged in LDS.
__global__ void LEAP_ctx(const float* __restrict__ a, const float* __restrict__ ctx,
                         float* __restrict__ c) {
    __shared__ float sa[NCTX];
    const int tid = threadIdx.x;
    for (int i = tid; i < NCTX; i += 128) sa[i] = a[i];
    __syncthreads();
    const int j = blockIdx.x * 128 + tid;
    float acc = 0.f;
#pragma unroll 4
    for (int k = 0; k < NCTX; ++k) acc += sa[k] * ctx[(size_t)k * E + j];
    c[j] = acc;
}

// gi = W_ih @ [c;x_t] + b_ih ; gh = W_hh @ h + b_hh. Wave-per-row GEMV (wave32),
// u and h staged in LDS, weight rows streamed from L2 with prefetch hints.
__global__ void LEAP_gates(const float* __restrict__ c, const float* __restrict__ dec_x,
                           const float* __restrict__ h, const float* __restrict__ w_ih,
                           const float* __restrict__ w_hh, const float* __restrict__ b_ih,
                           const float* __restrict__ b_hh, float* __restrict__ gi,
                           float* __restrict__ gh, int t) {
    __shared__ float su[2 * E];
    __shared__ float shh[E];
    const int tid = threadIdx.x;
    const float* x = dec_x + (size_t)t * E;
    for (int i = tid; i < E; i += 256) { su[i] = c[i]; su[E + i] = x[i]; shh[i] = h[i]; }
    const int wv = tid >> 5, lane = tid & 31;
    const int row = blockIdx.x * 8 + wv;            // 384 blocks * 8 waves = 3072 rows
    const float* wi = w_ih + (size_t)row * (2 * E);
    const float* wh = w_hh + (size_t)row * E;
    __builtin_prefetch(wi, 0, 1);
    __builtin_prefetch(wh, 0, 1);
    __syncthreads();
    float a1 = 0.f;
#pragma unroll
    for (int i = 0; i < 16; ++i) {
        const int idx = i * 128 + lane * 4;
        const float4 w4 = *(const float4*)(wi + idx);
        a1 += w4.x * su[idx] + w4.y * su[idx + 1] + w4.z * su[idx + 2] + w4.w * su[idx + 3];
    }
    float a2 = 0.f;
#pragma unroll
    for (int i = 0; i < 8; ++i) {
        const int idx = i * 128 + lane * 4;
        const float4 w4 = *(const float4*)(wh + idx);
        a2 += w4.x * shh[idx] + w4.y * shh[idx + 1] + w4.z * shh[idx + 2] + w4.w * shh[idx + 3];
    }
    a1 = waveSum(a1);
    a2 = waveSum(a2);
    if (lane == 0) { gi[row] = a1 + b_ih[row]; gh[row] = a2 + b_hh[row]; }
}

// Deferred output projection as one f32 WMMA GEMM:
// logits (65 x 32002) = Hrelu (80pad x 1024) @ out_w^T (1024 x 32002) + out_b.
// One wave per 16-wide N tile; all 5 M tiles accumulated so out_w streams once.
__global__ void LEAP_out_gemm(const float* __restrict__ Hrelu, const float* __restrict__ out_w,
                              const float* __restrict__ out_b, float* __restrict__ out) {
    const int lane = threadIdx.x;          // 32 threads (one wave)
    const int laneM = lane & 15;
    const int laneH = lane >> 4;           // 0 or 1
    const int n0 = blockIdx.x * 16;
    const int n = n0 + laneM;
    const int nrow = (n < VOUT) ? n : (VOUT - 1);
    const float* bptr = out_w + (size_t)nrow * E;
    const int koff = 2 * laneH;            // A: lanes 0-15 K=k,k+1 ; 16-31 K=k+2,k+3
    v8f acc[5] = {};
    for (int k = 0; k < E; k += 4) {
        const v2f bf = *(const v2f*)(bptr + k + koff);   // B[K][N]=out_w[N][K], row-striped
#pragma unroll
        for (int mt = 0; mt < 5; ++mt) {
            const v2f af = *(const v2f*)(Hrelu + (size_t)(mt * 16 + laneM) * E + k + koff);
            acc[mt] = __builtin_amdgcn_wmma_f32_16x16x4_f32(
                false, af, false, bf, (short)0, acc[mt], false, false);
        }
    }
    const float bias = out_b[nrow];
#pragma unroll
    for (int mt = 0; mt < 5; ++mt) {
#pragma unroll
        for (int v = 0; v < 8; ++v) {
            const int m = mt * 16 + v + 8 * laneH;   // C/D: VGPR v -> M=v / v+8
            if (m < TSTEPS && n < VOUT)
                out[(size_t)m * VOUT + n] = acc[mt][v] + bias;
        }
    }
}

extern "C" void kernel_launch(void* const* d_in, const int* in_sizes, int n_in,
                              void* d_out, int out_size, void* d_ws, size_t ws_size,
                              hipStream_t stream) {
    (void)in_sizes; (void)n_in; (void)out_size; (void)ws_size;
    const int* diag = (const int*)d_in[0];
    const int* proc = (const int*)d_in[1];
    const int* med  = (const int*)d_in[2];
    const int* dtok = (const int*)d_in[3];
    const float* enc_emb = (const float*)d_in[4];
    const float* dec_emb = (const float*)d_in[5];
    const float* attn_w  = (const float*)d_in[6];
    const float* attn_b  = (const float*)d_in[7];
    const float* w_ih = (const float*)d_in[8];
    const float* w_hh = (const float*)d_in[9];
    const float* b_ih = (const float*)d_in[10];
    const float* b_hh = (const float*)d_in[11];
    const float* out_w = (const float*)d_in[12];
    const float* out_b = (const float*)d_in[13];

    float* ws = (float*)d_ws;
    float* ctx       = ws;                      // 320*1024
    float* dec_x     = ctx + NCTX * E;          // 65*1024
    float* ctx_score = dec_x + TSTEPS * E;      // 320
    float* a         = ctx_score + NCTX;        // 320
    float* c         = a + NCTX;                // 1024
    float* h         = c + E;                   // 1024
    float* gi        = h + E;                   // 3072
    float* gh        = gi + 3 * E;              // 3072
    float* Hrelu     = gh + 3 * E;              // 80*1024 (rows 65..79 zeroed)

    LEAP_setup<<<NCTX + TSTEPS + (MPAD - TSTEPS), 256, 0, stream>>>(
        diag, proc, med, dtok, enc_emb, dec_emb, attn_w, attn_b,
        ctx, ctx_score, dec_x, h, Hrelu);
    LEAP_combine_attn<<<1, 1024, 0, stream>>>(gi, gh, h, attn_w, ctx_score, a, Hrelu, 0, 1);
    for (int t = 0; t < TSTEPS; ++t) {
        LEAP_ctx<<<8, 128, 0, stream>>>(a, ctx, c);
        LEAP_gates<<<384, 256, 0, stream>>>(c, dec_x, h, w_ih, w_hh, b_ih, b_hh, gi, gh, t);
        LEAP_combine_attn<<<1, 1024, 0, stream>>>(gi, gh, h, attn_w, ctx_score, a, Hrelu, t, 0);
    }
    LEAP_out_gemm<<<(VOUT + 15) / 16, 32, 0, stream>>>(Hrelu, out_w, out_b, (float*)d_out);
}